// SAGE_82386062671994
// MI455X (gfx1250) — compile-verified
//
#include <hip/hip_runtime.h>

typedef __attribute__((ext_vector_type(2))) float v2f;
typedef __attribute__((ext_vector_type(8))) float v8f;
typedef unsigned int v4u_ __attribute__((ext_vector_type(4)));
typedef int v4i_ __attribute__((ext_vector_type(4)));
typedef int v8i_ __attribute__((ext_vector_type(8)));

#define N_NODES 100000
#define CH 64

// ---------------------------------------------------------------- utilities
__global__ void zero_f32_kernel(float* __restrict__ p, int n) {
    int i = blockIdx.x * 256 + threadIdx.x;
    if (i < n) p[i] = 0.f;
}

__global__ void degree_kernel(const int* __restrict__ dst, float* __restrict__ deg, int E) {
    int e = blockIdx.x * 256 + threadIdx.x;
    if (e < E) atomicAdd(&deg[dst[e]], 1.0f);
}

__global__ void invdeg_kernel(const float* __restrict__ deg, float* __restrict__ inv, int n) {
    int i = blockIdx.x * 256 + threadIdx.x;
    if (i < n) {
        float d = deg[i];
        inv[i] = (d > 0.f) ? (1.0f / d) : 0.f;
    }
}

// ------------------------------------------------- edge gather + scatter-add
// thread = (edge, 4-channel chunk). Lanes 0..15 of a wave cover one edge's
// 64 channels -> coalesced 256B gather of feat[src]; f32 atomics into agg
// resolve in L2 (25.6MB node array is L2-resident on a 192MB L2).
__global__ void scatter_kernel(const float* __restrict__ feat,
                               const int* __restrict__ src,
                               const int* __restrict__ dst,
                               float* __restrict__ agg, int E) {
    int tid = blockIdx.x * 256 + threadIdx.x;
    int e = tid >> 4;
    if (e >= E) return;
    int c4 = (tid & 15) * 4;
    int s = src[e], d = dst[e];
    const float4 v = *(const float4*)(feat + (long)s * CH + c4);
    float* a = agg + (long)d * CH + c4;
    atomicAdd(a + 0, v.x);
    atomicAdd(a + 1, v.y);
    atomicAdd(a + 2, v.z);
    atomicAdd(a + 3, v.w);
}

// --------------------------------------------------------- TDM tile stage
// DMA a tile of up to 16 rows x 64 f32 (row stride 64 f32 in memory) into LDS
// with hardware padding: after every 256B row, pad 2 DWORDs -> LDS pitch 66
// floats (bank-conflict-free fragment reads). rows beyond tensor_dim1 are
// zero-filled by TDM OOB handling (ragged last block).
__device__ __forceinline__ void tdm_load_tile_f32(unsigned lds_byte,
                                                  const float* gptr,
                                                  int rows_avail) {
    if (rows_avail < 0) rows_avail = 0;
    unsigned long long ga = (unsigned long long)(uintptr_t)gptr;

    // D# group 0: count=1 | lds_addr | global_addr(57b) | type=2
    unsigned w0 = 1u;
    unsigned w1 = (unsigned)__builtin_amdgcn_readfirstlane((int)lds_byte);
    unsigned w2 = (unsigned)__builtin_amdgcn_readfirstlane((int)(unsigned)(ga & 0xffffffffull));
    unsigned w3 = (unsigned)__builtin_amdgcn_readfirstlane(
        (int)((((unsigned)(ga >> 32)) & 0x01ffffffu) | (2u << 30)));
    unsigned ra = (unsigned)__builtin_amdgcn_readfirstlane(rows_avail);

    // D# group 1 (packed as 4 x u64):
    //  data_size=4B(2), pad_enable, pad_interval=64 DWORDs(5), pad_amount=2 DWORDs(1)
    //  tensor_dim0=64, tensor_dim1=rows_avail, tile_dim0=64, tile_dim1=16,
    //  tensor_dim0_stride=64, tensor_dim1_stride unused.
    unsigned long long q0 = (2ull << 16) | (1ull << 20) | (5ull << 22) | (1ull << 25)
                          | (64ull << 48);
    unsigned long long q1 = ((unsigned long long)ra << 16) | (64ull << 48);
    unsigned long long q2 = 16ull | (64ull << 32);
    unsigned long long q3 = 0ull;

    v4u_ g0 = {w0, w1, w2, w3};
    v8i_ g1 = {(int)(q0 & 0xffffffffull), (int)(q0 >> 32),
               (int)(q1 & 0xffffffffull), (int)(q1 >> 32),
               (int)(q2 & 0xffffffffull), (int)(q2 >> 32),
               (int)(q3 & 0xffffffffull), (int)(q3 >> 32)};
    v4i_ gz = {0, 0, 0, 0};
#if defined(__clang_major__) && __clang_major__ >= 23
    v8i_ gz8 = {0, 0, 0, 0, 0, 0, 0, 0};
    __builtin_amdgcn_tensor_load_to_lds(g0, g1, gz, gz, gz8, 0);
#else
    __builtin_amdgcn_tensor_load_to_lds(g0, g1, gz, gz, 0);
#endif
}

// ------------------------------------------------------- fused WMMA SAGE GEMM
// out[m,n] = sum_k agg[m,k]*inv_deg[m]*Wl[n,k] + bl[n] + sum_k x[m,k]*Wr[n,k]
// One wave -> 16 rows (TDM-staged in LDS); block of 8 waves -> 128 rows.
template <int NOUT, bool RELU>
__global__ __launch_bounds__(256)
void sage_gemm_kernel(const float* __restrict__ agg,
                      const float* __restrict__ xin,
                      const float* __restrict__ inv_deg,
                      const float* __restrict__ Wl,
                      const float* __restrict__ bl,
                      const float* __restrict__ Wr,
                      float* __restrict__ out, int M) {
    constexpr int NTILES = (NOUT + 15) / 16;
    constexpr int NPAD = NTILES * 16;
    constexpr int WS = 66;        // LDS pitch in floats (256B row + 8B pad)
    constexpr int TILE = 16 * WS; // floats per staged 16x64 A tile
    __shared__ float lWl[NPAD * WS];
    __shared__ float lWr[NPAD * WS];
    __shared__ float lA[8 * 2 * TILE]; // per-wave {agg, x} tiles

    // weights -> LDS (cooperative, padded pitch, zero-pad rows >= NOUT)
    for (int i = threadIdx.x; i < NPAD * CH; i += 256) {
        int n = i >> 6, k = i & 63;
        lWl[n * WS + k] = (n < NOUT) ? Wl[n * CH + k] : 0.f;
        lWr[n * WS + k] = (n < NOUT) ? Wr[n * CH + k] : 0.f;
    }

    const int wave = __builtin_amdgcn_readfirstlane(threadIdx.x >> 5);
    const int lane = threadIdx.x & 31;
    const int half = lane >> 4;   // K-pair select for A/B fragments
    const int l16  = lane & 15;
    const int row0 = blockIdx.x * 128 + wave * 16;

    // async DMA of this wave's two 16x64 slabs into LDS (zero-filled OOB rows)
    float* tA = &lA[wave * 2 * TILE];
    float* tX = tA + TILE;
    tdm_load_tile_f32((unsigned)(uintptr_t)tA, agg + (size_t)row0 * CH, M - row0);
    tdm_load_tile_f32((unsigned)(uintptr_t)tX, xin + (size_t)row0 * CH, M - row0);
    __builtin_amdgcn_s_wait_tensorcnt(0);
    __syncthreads();  // publish weights (all waves) + order LDS reads after TDM

    const int rC = (row0 + l16 < M) ? (row0 + l16) : (M - 1);
    const float scale = inv_deg[rC];  // OOB rows are zero in LDS, so scale*0=0

    // C/D layout: lane's N index is constant across its 8 accumulator regs.
    v8f acc[NTILES];
#pragma unroll
    for (int t = 0; t < NTILES; ++t) {
        int n = t * 16 + l16;
        float b = (n < NOUT) ? bl[n] : 0.f;
#pragma unroll
        for (int j = 0; j < 8; ++j) acc[t][j] = b;
    }

    // K=64 in 16 steps of 4; branch-free, EXEC all-ones (WMMA requirement).
#pragma unroll 4
    for (int s = 0; s < 16; ++s) {
        const int kk = s * 4 + half * 2;  // A 16x4 f32: lanes<16 K={0,1}, lanes>=16 K={2,3}
        const int aoff = l16 * WS + kk;
        v2f a_agg, a_x;
        a_agg[0] = tA[aoff] * scale;
        a_agg[1] = tA[aoff + 1] * scale;
        a_x[0] = tX[aoff];
        a_x[1] = tX[aoff + 1];
#pragma unroll
        for (int t = 0; t < NTILES; ++t) {
            const int n = t * 16 + l16;
            v2f b_l, b_r;
            b_l[0] = lWl[n * WS + kk];
            b_l[1] = lWl[n * WS + kk + 1];
            b_r[0] = lWr[n * WS + kk];
            b_r[1] = lWr[n * WS + kk + 1];
            acc[t] = __builtin_amdgcn_wmma_f32_16x16x4_f32(
                false, a_agg, false, b_l, (short)0, acc[t], false, false);
            acc[t] = __builtin_amdgcn_wmma_f32_16x16x4_f32(
                false, a_x, false, b_r, (short)0, acc[t], false, false);
        }
    }

    // D layout: lane half=0 holds M=0..7 in regs 0..7, half=1 holds M=8..15.
#pragma unroll
    for (int t = 0; t < NTILES; ++t) {
        int n = t * 16 + l16;
        if (n >= NOUT) continue;
#pragma unroll
        for (int v = 0; v < 8; ++v) {
            int m = row0 + v + half * 8;
            if (m < M) {
                float r = acc[t][v];
                if (RELU) r = fmaxf(r, 0.f);
                out[(long)m * NOUT + n] = r;
            }
        }
    }
}

// ---------------------------------------------------------------- launcher
extern "C" void kernel_launch(void* const* d_in, const int* in_sizes, int n_in,
                              void* d_out, int out_size, void* d_ws, size_t ws_size,
                              hipStream_t stream) {
    const float* x   = (const float*)d_in[0];
    const int*   ei  = (const int*)d_in[1];
    const float* W1l = (const float*)d_in[2];
    const float* b1l = (const float*)d_in[3];
    const float* W1r = (const float*)d_in[4];
    const float* W2l = (const float*)d_in[5];
    const float* b2l = (const float*)d_in[6];
    const float* W2r = (const float*)d_in[7];
    const float* W3l = (const float*)d_in[8];
    const float* b3l = (const float*)d_in[9];
    const float* W3r = (const float*)d_in[10];

    const int N = N_NODES;
    const int E = in_sizes[1] / 2;
    const int* src = ei;
    const int* dst = ei + E;

    // workspace partition (floats)
    float* deg     = (float*)d_ws;           // N
    float* inv_deg = deg + N;                // N
    float* agg     = inv_deg + N;            // N*CH
    float* h1      = agg + (size_t)N * CH;   // N*CH
    float* h2      = h1 + (size_t)N * CH;    // N*CH
    float* out     = (float*)d_out;          // N*2

    const int ZB_N  = (N + 255) / 256;
    const int ZB_NC = (N * CH + 255) / 256;
    const int EB    = (E + 255) / 256;
    const int SB    = (E * 16 + 255) / 256;
    const int GB    = (N + 127) / 128;

    // degrees (recomputed every call: launch must be stateless/deterministic)
    zero_f32_kernel<<<ZB_N, 256, 0, stream>>>(deg, N);
    degree_kernel<<<EB, 256, 0, stream>>>(dst, deg, E);
    invdeg_kernel<<<ZB_N, 256, 0, stream>>>(deg, inv_deg, N);

    // layer 1: x -> h1
    zero_f32_kernel<<<ZB_NC, 256, 0, stream>>>(agg, N * CH);
    scatter_kernel<<<SB, 256, 0, stream>>>(x, src, dst, agg, E);
    sage_gemm_kernel<64, true><<<GB, 256, 0, stream>>>(agg, x, inv_deg, W1l, b1l, W1r, h1, N);

    // layer 2: h1 -> h2
    zero_f32_kernel<<<ZB_NC, 256, 0, stream>>>(agg, N * CH);
    scatter_kernel<<<SB, 256, 0, stream>>>(h1, src, dst, agg, E);
    sage_gemm_kernel<64, true><<<GB, 256, 0, stream>>>(agg, h1, inv_deg, W2l, b2l, W2r, h2, N);

    // layer 3: h2 -> out (2 channels via one zero-padded N-tile, no relu)
    zero_f32_kernel<<<ZB_NC, 256, 0, stream>>>(agg, N * CH);
    scatter_kernel<<<SB, 256, 0, stream>>>(h2, src, dst, agg, E);
    sage_gemm_kernel<2, false><<<GB, 256, 0, stream>>>(agg, h2, inv_deg, W3l, b3l, W3r, out, N);
}